// YOLOLoss_63273458205160
// MI455X (gfx1250) — compile-verified
//
#include <hip/hip_runtime.h>
#include <hip/hip_bf16.h>
#include <math.h>

// Problem constants (match reference: B, T, GRID, C = 128, 100, 52, 80)
#define Bn 128
#define Tn 100
#define Sn 52
#define Cn 80
#define Dn 85   // 5 + C channels, innermost dim

typedef __attribute__((ext_vector_type(2))) float v2f;
typedef __attribute__((ext_vector_type(8))) float v8f;

// total float4 count = B*S*S*85/4 = 7,354,880 = 2^9 * 5 * 13^2 * 17.
// 1690 blocks * 256 threads = 432,640 lanes -> exactly 17 iterations, no tail.
#define K1_BLOCKS 1690
#define K1_THREADS 256
#define K1_ITERS 17
#define K2_THREADS 256
#define K2_BLOCKS ((Bn * Tn) / K2_THREADS)   // 12800 / 256 = 50 exactly

static_assert((long)Bn * Sn * Sn * Dn / 4 == (long)K1_BLOCKS * K1_THREADS * K1_ITERS,
              "grid must tile the tensor exactly");

// jax.nn.softplus(z) = max(z,0) + log1p(exp(-|z|)).
// Hardware trans version: t = e^{-|z|} in (0,1], so log(1+t) is evaluated on
// (1,2] where v_log_f32 is well conditioned. ~1e-6 rel error on a mean loss.
__device__ __forceinline__ float softplusf(float z) {
    return fmaxf(z, 0.0f) + __logf(1.0f + __expf(-fabsf(z)));
}

// Kernel 1: streaming pass over all of p. Sums softplus over obj channel
// (scalar acc) and cls channels (WMMA f32 16x16x4 accumulation: D = 1*B + C,
// i.e. per-column running sums in the C matrix; exact f32 math).
__global__ __launch_bounds__(K1_THREADS)
void k1_softplus_sums(const float* __restrict__ p,
                      float* __restrict__ part_cls,
                      float* __restrict__ part_obj) {
    const long stride = (long)K1_BLOCKS * K1_THREADS;   // in float4 units
    long i = (long)blockIdx.x * K1_THREADS + threadIdx.x;

    v2f ones; ones[0] = 1.0f; ones[1] = 1.0f;           // A = ones(16x4)
    v8f acc0 = {0.f,0.f,0.f,0.f,0.f,0.f,0.f,0.f};
    v8f acc1 = {0.f,0.f,0.f,0.f,0.f,0.f,0.f,0.f};
    float obj_acc = 0.0f;

    // channel of first float of this thread's float4; advance incrementally
    int ch = (int)((i * 4) % Dn);
    const int chstep = (int)((stride * 4) % Dn);        // = 45

    const float4* __restrict__ p4 = (const float4*)p;

    for (int it = 0; it < K1_ITERS; ++it, i += stride) {
        const float4 v = p4[i];   // always in range; EXEC stays all-ones

        const int c0i = ch;
        int c1i = ch + 1; if (c1i >= Dn) c1i -= Dn;
        int c2i = ch + 2; if (c2i >= Dn) c2i -= Dn;
        int c3i = ch + 3; if (c3i >= Dn) c3i -= Dn;

        // Branch-free: softplus everything, route via selects.
        const float s0 = softplusf(v.x);
        const float s1 = softplusf(v.y);
        const float s2 = softplusf(v.z);
        const float s3 = softplusf(v.w);

        obj_acc += (c0i == 4) ? s0 : 0.f;
        obj_acc += (c1i == 4) ? s1 : 0.f;
        obj_acc += (c2i == 4) ? s2 : 0.f;
        obj_acc += (c3i == 4) ? s3 : 0.f;

        v2f b01; b01[0] = (c0i >= 5) ? s0 : 0.f; b01[1] = (c1i >= 5) ? s1 : 0.f;
        v2f b23; b23[0] = (c2i >= 5) ? s2 : 0.f; b23[1] = (c3i >= 5) ? s3 : 0.f;

        // Two independent accumulator chains to pipeline WMMA issue.
        acc0 = __builtin_amdgcn_wmma_f32_16x16x4_f32(
                   false, ones, false, b01, (short)0, acc0, false, false);
        acc1 = __builtin_amdgcn_wmma_f32_16x16x4_f32(
                   false, ones, false, b23, (short)0, acc1, false, false);

        ch += chstep; if (ch >= Dn) ch -= Dn;
    }

    // D rows are identical (A == ones), so VGPR0 holds colsum[lane & 15].
    float wave_cls = acc0[0] + acc1[0];
    #pragma unroll
    for (int m = 1; m <= 8; m <<= 1) wave_cls += __shfl_xor(wave_cls, m, 32);
    float wave_obj = obj_acc;
    #pragma unroll
    for (int m = 1; m <= 16; m <<= 1) wave_obj += __shfl_xor(wave_obj, m, 32);

    __shared__ float s_cls[K1_THREADS / 32];
    __shared__ float s_obj[K1_THREADS / 32];
    const int wave = threadIdx.x >> 5, lane = threadIdx.x & 31;
    if (lane == 0) { s_cls[wave] = wave_cls; s_obj[wave] = wave_obj; }
    __syncthreads();
    if (threadIdx.x == 0) {
        float tc = 0.f, to = 0.f;
        #pragma unroll
        for (int w = 0; w < K1_THREADS / 32; ++w) { tc += s_cls[w]; to += s_obj[w]; }
        part_cls[blockIdx.x] = tc;
        part_obj[blockIdx.x] = to;
    }
}

// Kernel 2: sparse corrections. One thread per (b, t). Dedup via
// "last occurrence wins" (matches XLA scatter ordering):
//   obj dot:  z_obj at each unique target cell
//   cls dot:  z_cls at each unique (cell, class) pair
//   mse:      sum (p - box_last)^2 over 4 channels at each unique cell
__global__ __launch_bounds__(K2_THREADS)
void k2_corrections(const float* __restrict__ p,
                    const float* __restrict__ tb,
                    const int* __restrict__ tc,
                    float* __restrict__ part_cobj,
                    float* __restrict__ part_ccls,
                    float* __restrict__ part_cmse) {
    const int tid = blockIdx.x * blockDim.x + threadIdx.x;
    float cobj = 0.f, ccls = 0.f, cmse = 0.f;
    if (tid < Bn * Tn) {
        const int b = tid / Tn, t = tid % Tn;
        const float* __restrict__ tbb = tb + (long)b * Tn * 4;
        const int* __restrict__ tcb = tc + b * Tn;

        const float x = tbb[t * 4 + 0], y = tbb[t * 4 + 1];
        int gx = (int)(x * (float)Sn); gx = gx < 0 ? 0 : (gx > Sn - 1 ? Sn - 1 : gx);
        int gy = (int)(y * (float)Sn); gy = gy < 0 ? 0 : (gy > Sn - 1 ? Sn - 1 : gy);
        const int cls = tcb[t];

        bool last_cell = true, last_pair = true;
        for (int u = t + 1; u < Tn; ++u) {
            const float xu = tbb[u * 4 + 0], yu = tbb[u * 4 + 1];
            int gxu = (int)(xu * (float)Sn); gxu = gxu < 0 ? 0 : (gxu > Sn - 1 ? Sn - 1 : gxu);
            int gyu = (int)(yu * (float)Sn); gyu = gyu < 0 ? 0 : (gyu > Sn - 1 ? Sn - 1 : gyu);
            if (gxu == gx && gyu == gy) {
                last_cell = false;
                if (tcb[u] == cls) last_pair = false;
            }
        }

        const float* __restrict__ cell = p + (((long)b * Sn + gy) * Sn + gx) * Dn;
        if (last_pair) ccls = cell[5 + cls];
        if (last_cell) {
            cobj = cell[4];
            #pragma unroll
            for (int c2 = 0; c2 < 4; ++c2) {
                const float d = cell[c2] - tbb[t * 4 + c2];
                cmse += d * d;
            }
        }
    }

    #pragma unroll
    for (int m = 1; m <= 16; m <<= 1) {
        cobj += __shfl_xor(cobj, m, 32);
        ccls += __shfl_xor(ccls, m, 32);
        cmse += __shfl_xor(cmse, m, 32);
    }
    __shared__ float so[K2_THREADS / 32], sc[K2_THREADS / 32], sm[K2_THREADS / 32];
    const int wave = threadIdx.x >> 5, lane = threadIdx.x & 31;
    if (lane == 0) { so[wave] = cobj; sc[wave] = ccls; sm[wave] = cmse; }
    __syncthreads();
    if (threadIdx.x == 0) {
        float a = 0.f, bsum = 0.f, csum = 0.f;
        #pragma unroll
        for (int w = 0; w < K2_THREADS / 32; ++w) { a += so[w]; bsum += sc[w]; csum += sm[w]; }
        part_cobj[blockIdx.x] = a;
        part_ccls[blockIdx.x] = bsum;
        part_cmse[blockIdx.x] = csum;
    }
}

// Kernel 3: deterministic final combine (single block).
__global__ __launch_bounds__(256)
void k3_final(const float* __restrict__ part_cls, const float* __restrict__ part_obj,
              int nb1,
              const float* __restrict__ part_cobj, const float* __restrict__ part_ccls,
              const float* __restrict__ part_cmse, int nb2,
              float* __restrict__ out) {
    __shared__ float red[5][256];
    float v0 = 0.f, v1 = 0.f, v2 = 0.f, v3 = 0.f, v4 = 0.f;
    for (int i = threadIdx.x; i < nb1; i += blockDim.x) { v0 += part_cls[i]; v1 += part_obj[i]; }
    for (int i = threadIdx.x; i < nb2; i += blockDim.x) { v2 += part_cobj[i]; v3 += part_ccls[i]; v4 += part_cmse[i]; }
    red[0][threadIdx.x] = v0; red[1][threadIdx.x] = v1; red[2][threadIdx.x] = v2;
    red[3][threadIdx.x] = v3; red[4][threadIdx.x] = v4;
    __syncthreads();
    for (int s = 128; s > 0; s >>= 1) {
        if ((int)threadIdx.x < s) {
            #pragma unroll
            for (int k = 0; k < 5; ++k)
                red[k][threadIdx.x] += red[k][threadIdx.x + s];
        }
        __syncthreads();
    }
    if (threadIdx.x == 0) {
        const float sp_cls = red[0][0];
        const float sp_obj = red[1][0];
        const float dot_obj = red[2][0];
        const float dot_cls = red[3][0];
        const float mse_sum = red[4][0];
        const float Nobj = (float)((long)Bn * Sn * Sn);            // 346112
        const float Ncls = (float)((long)Bn * Sn * Sn * Cn);       // 27688960
        const float Nmse = (float)((long)Bn * Sn * Sn * 4);        // 1384448
        out[0] = mse_sum / Nmse + (sp_obj - dot_obj) / Nobj + (sp_cls - dot_cls) / Ncls;
    }
}

extern "C" void kernel_launch(void* const* d_in, const int* in_sizes, int n_in,
                              void* d_out, int out_size, void* d_ws, size_t ws_size,
                              hipStream_t stream) {
    const float* p  = (const float*)d_in[0];   // [B,S,S,85] f32
    const float* tb = (const float*)d_in[1];   // [B,T,4]    f32
    const int*   tc = (const int*)d_in[2];     // [B,T]      i32
    // d_in[3] is the scalar S (=52), folded into compile-time constants.
    (void)in_sizes; (void)n_in; (void)out_size; (void)ws_size;

    float* ws = (float*)d_ws;
    float* part_cls  = ws;
    float* part_obj  = ws + K1_BLOCKS;
    float* part_cobj = ws + 2 * K1_BLOCKS;
    float* part_ccls = part_cobj + K2_BLOCKS;
    float* part_cmse = part_ccls + K2_BLOCKS;

    k1_softplus_sums<<<K1_BLOCKS, K1_THREADS, 0, stream>>>(p, part_cls, part_obj);
    k2_corrections<<<K2_BLOCKS, K2_THREADS, 0, stream>>>(p, tb, tc,
                                                         part_cobj, part_ccls, part_cmse);
    k3_final<<<1, 256, 0, stream>>>(part_cls, part_obj, K1_BLOCKS,
                                    part_cobj, part_ccls, part_cmse, K2_BLOCKS,
                                    (float*)d_out);
}